// FeatureDRO_25735444038246
// MI455X (gfx1250) — compile-verified
//
#include <hip/hip_runtime.h>

#define H 32
#define D 128
#define NPAIRS (H * (H - 1) / 2)   // 496
#define WAVES 4                    // waves per block in pass 1
#define SLOT 4160                  // per-wave LDS floats: 4096 acc + 32 cnt + pad
#define PACC (H * D + H)           // 4128 useful floats per partial
#define PSLOT 4160                 // partial stride (floats) in workspace
#define MEANS_OFF 4224             // floats reserved for means at ws[0]
#define MAXBLK 1024

typedef __attribute__((ext_vector_type(2))) float v2f;
typedef __attribute__((ext_vector_type(8))) float v8f;

// ---------------- Pass 1: per-wave private LDS segment sums (deterministic) --
__global__ void fdro_partial(const float* __restrict__ feat,
                             const int* __restrict__ ids,
                             int n, float* __restrict__ partials) {
  extern __shared__ float lds[];
  const int tid  = threadIdx.x;
  const int lane = tid & 31;
  const int wave = tid >> 5;
  float* wbase = lds + wave * SLOT;
  for (int i = lane; i < SLOT; i += 32) wbase[i] = 0.0f;  // zero own slot (DS in-order per wave)

  float4*       wacc  = (float4*)wbase;        // [H][32] float4
  float*        wcnt  = wbase + H * D;         // [H]
  const float4* feat4 = (const float4*)feat;   // row r = feat4[r*32 + lane]

  const int gw     = blockIdx.x * WAVES + wave;
  const int stride = gridDim.x * WAVES;

  int r = gw;
  for (; r + 3 * stride < n; r += 4 * stride) {
    const int r0 = r, r1 = r + stride, r2 = r + 2 * stride, r3 = r + 3 * stride;
    float4 x0 = feat4[(size_t)r0 * 32 + lane];
    float4 x1 = feat4[(size_t)r1 * 32 + lane];
    float4 x2 = feat4[(size_t)r2 * 32 + lane];
    float4 x3 = feat4[(size_t)r3 * 32 + lane];
    __builtin_prefetch(&feat4[(size_t)(r + 4 * stride) * 32 + lane], 0, 0);
    const int h0 = __builtin_amdgcn_readfirstlane(ids[r0]);
    const int h1 = __builtin_amdgcn_readfirstlane(ids[r1]);
    const int h2 = __builtin_amdgcn_readfirstlane(ids[r2]);
    const int h3 = __builtin_amdgcn_readfirstlane(ids[r3]);
    float4* p;  float4 t;
    p = &wacc[h0 * 32 + lane]; t = *p; t.x += x0.x; t.y += x0.y; t.z += x0.z; t.w += x0.w; *p = t;
    p = &wacc[h1 * 32 + lane]; t = *p; t.x += x1.x; t.y += x1.y; t.z += x1.z; t.w += x1.w; *p = t;
    p = &wacc[h2 * 32 + lane]; t = *p; t.x += x2.x; t.y += x2.y; t.z += x2.z; t.w += x2.w; *p = t;
    p = &wacc[h3 * 32 + lane]; t = *p; t.x += x3.x; t.y += x3.y; t.z += x3.z; t.w += x3.w; *p = t;
    if (lane == 0) { wcnt[h0] += 1.f; wcnt[h1] += 1.f; wcnt[h2] += 1.f; wcnt[h3] += 1.f; }
  }
  for (; r < n; r += stride) {
    float4 x = feat4[(size_t)r * 32 + lane];
    const int h = __builtin_amdgcn_readfirstlane(ids[r]);
    float4* p = &wacc[h * 32 + lane];
    float4 t = *p; t.x += x.x; t.y += x.y; t.z += x.z; t.w += x.w; *p = t;
    if (lane == 0) wcnt[h] += 1.f;
  }

  __syncthreads();
  // combine the 4 wave slots in fixed order -> one partial per block
  float* dst = partials + (size_t)blockIdx.x * PSLOT;
  for (int i = tid; i < PACC; i += blockDim.x)
    dst[i] = (lds[i] + lds[SLOT + i]) + (lds[2 * SLOT + i] + lds[3 * SLOT + i]);
}

// ---------------- Pass 2: deterministic reduce of partials -> means ---------
__global__ void fdro_reduce(const float* __restrict__ partials, int nblocks,
                            float* __restrict__ means) {
  __shared__ float csum[128];
  const int h = blockIdx.x;    // 0..31
  const int t = threadIdx.x;   // 0..127

  float c = 0.f;
  for (int b = t; b < nblocks; b += 128)
    c += partials[(size_t)b * PSLOT + H * D + h];
  csum[t] = c;
  __syncthreads();
  for (int s = 64; s > 0; s >>= 1) {
    if (t < s) csum[t] += csum[t + s];
    __syncthreads();
  }
  const float inv = 1.0f / csum[0];

  // one thread per (h, d) element, fixed 4-accumulator order
  const int d = t;
  float s0 = 0.f, s1 = 0.f, s2 = 0.f, s3 = 0.f;
  int b = 0;
  for (; b + 3 < nblocks; b += 4) {
    s0 += partials[(size_t)(b + 0) * PSLOT + h * D + d];
    s1 += partials[(size_t)(b + 1) * PSLOT + h * D + d];
    s2 += partials[(size_t)(b + 2) * PSLOT + h * D + d];
    s3 += partials[(size_t)(b + 3) * PSLOT + h * D + d];
  }
  for (; b < nblocks; ++b) s0 += partials[(size_t)b * PSLOT + h * D + d];
  means[h * D + d] = ((s0 + s1) + (s2 + s3)) * inv;
}

// ---------------- Pass 3: Gram matrix via WMMA f32, pairwise distances ------
__global__ void fdro_pairwise(const float* __restrict__ means_g,
                              float* __restrict__ out) {
  __shared__ float m[H * D];
  __shared__ float nrm[H];
  __shared__ float wsum[4];
  const int tid = threadIdx.x;  // 128 threads = 4 waves
  for (int i = tid; i < H * D; i += 128) m[i] = means_g[i];
  __syncthreads();
  if (tid < H) {
    float s = 0.f;
    for (int d = 0; d < D; ++d) { float x = m[tid * D + d]; s += x * x; }
    nrm[tid] = s;
  }
  __syncthreads();

  const int wave = tid >> 5, lane = tid & 31;
  const int tm = wave >> 1, tn = wave & 1;       // 16x16 tile of 32x32 Gram
  const int half = lane >> 4, rr = lane & 15;
  const float* arow = &m[(tm * 16 + rr) * D];    // A: rows of means (M x K)
  const float* brow = &m[(tn * 16 + rr) * D];    // B: means^T (K x N) -> same rows

  v8f c = {};
  for (int k = 0; k < D; k += 4) {
    const int kk = k + 2 * half;                 // VGPR0: K,K+2 ; VGPR1: K+1,K+3
    v2f a, b;
    a[0] = arow[kk]; a[1] = arow[kk + 1];
    b[0] = brow[kk]; b[1] = brow[kk + 1];
    c = __builtin_amdgcn_wmma_f32_16x16x4_f32(false, a, false, b, (short)0, c,
                                              false, false);
  }

  const int gj = tn * 16 + rr;                   // column index of this lane
  float s = 0.f;
#pragma unroll
  for (int v = 0; v < 8; ++v) {
    const int gi = tm * 16 + v + 8 * half;       // row index for C VGPR v
    if (gi < gj) {
      float sq = nrm[gi] + nrm[gj] - 2.0f * c[v];
      s += sqrtf(fmaxf(sq, 0.0f));
    }
  }
  for (int off = 16; off > 0; off >>= 1) s += __shfl_xor(s, off, 32);
  if (lane == 0) wsum[wave] = s;
  __syncthreads();
  if (tid == 0) out[0] = ((wsum[0] + wsum[1]) + (wsum[2] + wsum[3])) / (float)NPAIRS;
}

// ---------------------------------------------------------------------------
extern "C" void kernel_launch(void* const* d_in, const int* in_sizes, int n_in,
                              void* d_out, int out_size, void* d_ws, size_t ws_size,
                              hipStream_t stream) {
  const float* feat = (const float*)d_in[0];
  const int*   ids  = (const int*)d_in[1];
  const int    n    = in_sizes[1];

  float* ws       = (float*)d_ws;
  float* means    = ws;                 // MEANS_OFF floats reserved
  float* partials = ws + MEANS_OFF;

  long slots = ((long)(ws_size / 4) - MEANS_OFF) / PSLOT;
  if (slots < 1) slots = 1;
  if (slots > MAXBLK) slots = MAXBLK;
  const int nblocks = (int)slots;

  fdro_partial<<<nblocks, WAVES * 32, WAVES * SLOT * sizeof(float), stream>>>(
      feat, ids, n, partials);
  fdro_reduce<<<H, 128, 0, stream>>>(partials, nblocks, means);
  fdro_pairwise<<<1, 128, 0, stream>>>(means, (float*)d_out);
}